// ModNN_40553081209621
// MI455X (gfx1250) — compile-verified
//
#include <hip/hip_runtime.h>
#include <math.h>

#define B_DIM 2048
#define T_DIM 1440
#define NRC   5
#define ENC   48

typedef __attribute__((ext_vector_type(2))) float v2f;
typedef __attribute__((ext_vector_type(8))) float v8f;

__device__ __forceinline__ float sp(float x) {       // softplus (stable)
    return (x > 20.0f) ? x : log1pf(__expf(x));
}
__device__ __forceinline__ float sigm(float x) {
    return 1.0f / (1.0f + __expf(-x));
}

// ---------------------------------------------------------------------------
// Kernel 1: per-(b,t) MLP via WMMA + pack transposed [t][b] float4 stream
//   {t_amb, sol, q_int+q_hvac+q_dir, tz_gt_next}
// One wave = 16 rows (one 16x16x4 f32 WMMA pair). 8 waves / block.
// All lane-dependent loads are unconditional (addresses valid for every
// lane); K-padding zeros are applied with selects, not branches.
// ---------------------------------------------------------------------------
__global__ __launch_bounds__(256) void prep_kernel(
    const float* __restrict__ X,
    const float* __restrict__ W1, const float* __restrict__ b1,
    const float* __restrict__ W2, const float* __restrict__ b2,
    const float* __restrict__ hvac_gain, const float* __restrict__ int_gain,
    const float* __restrict__ direct_gain,
    float4* __restrict__ ws4)
{
    __shared__ float hsh[8][16 * 33];          // 16 rows x 32 cols, pad 33

    const int wv   = threadIdx.x >> 5;
    const int lane = threadIdx.x & 31;
    const int tile = blockIdx.x * 8 + wv;      // grid sized exactly
    const int row0 = tile * 16;                // T % 16 == 0 -> tile = one b
    const bool lo  = (lane < 16);

    // --- per-lane row loads, unconditional (row valid for all lanes)
    const int row = row0 + (lane & 15);
    const float* xr = X + (size_t)row * 7;
    const float ta = xr[1];
    const float so = xr[2];
    const float x0 = xr[3];
    const float x1 = xr[4];
    const float x2 = xr[5];
    const float hv = xr[6];
    const int   t    = row % T_DIM;
    const int   rown = (t + 1 < T_DIM) ? (row + 1) : row;
    const float gtn  = X[(size_t)rown * 7 + 0];

    // --- A: 16x4 f32 (K=2 real; lanes 16..31 carry K=2,3 -> zero via select)
    v2f a;
    a.x = lo ? x0 : 0.f;
    a.y = lo ? x1 : 0.f;

    // --- B: 4x16 tiles of W1^T (h cols 0..15 and 16..31)
    const int n = lane & 15;
    const float w00 = W1[n * 2 + 0],        w01 = W1[n * 2 + 1];
    const float w10 = W1[(n + 16) * 2 + 0], w11 = W1[(n + 16) * 2 + 1];
    v2f bm0, bm1;
    bm0.x = lo ? w00 : 0.f;  bm0.y = lo ? w01 : 0.f;
    bm1.x = lo ? w10 : 0.f;  bm1.y = lo ? w11 : 0.f;

    v8f c = {};
    v8f d0 = __builtin_amdgcn_wmma_f32_16x16x4_f32(false, a, false, bm0,
                                                   (short)0, c, false, false);
    v8f d1 = __builtin_amdgcn_wmma_f32_16x16x4_f32(false, a, false, bm1,
                                                   (short)0, c, false, false);

    // --- bias + relu, scatter D layout into LDS as dense 16x32
    const float bn0 = b1[n], bn1 = b1[n + 16];
    float* hbase = &hsh[wv][0];
    const int mofs = lo ? 0 : 8;
#pragma unroll
    for (int r = 0; r < 8; ++r) {
        const int m = r + mofs;
        hbase[m * 33 + n]      = fmaxf(d0[r] + bn0, 0.f);
        hbase[m * 33 + n + 16] = fmaxf(d1[r] + bn1, 0.f);
    }
    __syncthreads();

    // --- layer 2 dot (32 FMA) + pack transposed stream
    if (lo) {
        float y = b2[0];
        const float* hr = &hsh[wv][n * 33];
#pragma unroll
        for (int k = 0; k < 32; ++k) y = fmaf(W2[k], hr[k], y);
        const float sch = sigm(y) + x2;
        const float q = sp(int_gain[0])    * 0.1f * sch
                      + sp(hvac_gain[0])   * 0.1f * hv
                      + sp(direct_gain[0]) * 0.5f * so;
        const int b = row / T_DIM;
        ws4[(size_t)t * B_DIM + b] = make_float4(ta, so, q, gtn);
    }
}

// ---------------------------------------------------------------------------
// Kernel 2: sequential scan, 1 thread per batch element.
// float4 loads are wave-coalesced (consecutive b -> one b128/line) and hit L2
// (47 MB stream << 192 MB L2). Prefetch 8 steps ahead to cover L2 latency at
// low occupancy. Outputs staged through LDS for coalesced 128B stores.
// ---------------------------------------------------------------------------
__global__ __launch_bounds__(128) void scan_kernel(
    const float* __restrict__ X,
    const float* __restrict__ rcR, const float* __restrict__ rcC,
    const float* __restrict__ winR,
    const float* __restrict__ abs_wall, const float* __restrict__ abs_roof,
    const float* __restrict__ zone_C_inv,
    const float4* __restrict__ ws4,
    float* __restrict__ out)
{
    __shared__ float tzs[128 * 33];            // 128 b x 32 t transpose tile

    const int tid = threadIdx.x;
    const int b0  = blockIdx.x * 128;
    const int b   = b0 + tid;
    const int wv  = tid >> 5, lane = tid & 31;

    // fold constants once per thread
    float r_inv[NRC], cdt[NRC], arc[NRC];
    const float saw = sp(abs_wall[0]) * 0.5f;
    const float sar = sp(abs_roof[0]) * 0.5f;
#pragma unroll
    for (int j = 0; j < NRC; ++j) {
        r_inv[j] = sp(rcR[j]) * 0.1f;
        cdt[j]   = sp(rcC[j]) * (1.0f / 100000.0f) * 900.0f;
        arc[j]   = (j < 4) ? saw : sar;
    }
    const float wr   = (sp(winR[0]) + sp(winR[1])) * 0.5f;
    const float czdt = sp(zone_C_inv[0]) * (1.0f / 100000.0f) * 900.0f;

    // initial state
    float Tz  = X[(size_t)b * T_DIM * 7 + 0];
    const float ta0 = X[(size_t)b * T_DIM * 7 + 1];
    float Tmid[NRC];
#pragma unroll
    for (int j = 0; j < NRC; ++j) Tmid[j] = 0.7f * Tz + 0.3f * ta0;

    for (int tc = 0; tc < T_DIM; tc += 32) {
#pragma unroll 4
        for (int k = 0; k < 32; ++k) {
            const int t = tc + k;
            // warm the WGP cache 8 steps ahead (global_prefetch_b8)
            if (t + 8 < T_DIM)
                __builtin_prefetch(&ws4[(size_t)(t + 8) * B_DIM + b], 0, 0);

            const float4 ld = ws4[(size_t)t * B_DIM + b];
            const float ta = ld.x, so = ld.y, q = ld.z, gtn = ld.w;

            float qosum = 0.f;
#pragma unroll
            for (int j = 0; j < NRC; ++j) {
                qosum += (Tmid[j] - Tz) * r_inv[j];
                const float dm = (ta + Tz - 2.0f * Tmid[j]) * r_inv[j]
                               + so * arc[j];
                Tmid[j] = fmaf(dm, cdt[j], Tmid[j]);
            }
            const float Tz_new = fmaf(qosum + (ta - Tz) * wr + q, czdt, Tz);
            tzs[tid * 33 + k] = Tz_new;
            Tz = (t + 1 < ENC) ? gtn : Tz_new;
        }
        __syncthreads();
        // coalesced writeout: each wave drains 32 rows, 128B lines along t
#pragma unroll 4
        for (int i = 0; i < 32; ++i) {
            const int rb = wv * 32 + i;
            out[(size_t)(b0 + rb) * T_DIM + tc + lane] = tzs[rb * 33 + lane];
        }
        __syncthreads();
    }
}

// ---------------------------------------------------------------------------
extern "C" void kernel_launch(void* const* d_in, const int* in_sizes, int n_in,
                              void* d_out, int out_size, void* d_ws, size_t ws_size,
                              hipStream_t stream)
{
    const float* X    = (const float*)d_in[0];
    const float* rcR  = (const float*)d_in[1];
    const float* rcC  = (const float*)d_in[2];
    const float* winR = (const float*)d_in[3];
    const float* hvac = (const float*)d_in[4];
    const float* W1   = (const float*)d_in[5];
    const float* b1   = (const float*)d_in[6];
    const float* W2   = (const float*)d_in[7];
    const float* b2   = (const float*)d_in[8];
    const float* ig   = (const float*)d_in[9];
    const float* dg   = (const float*)d_in[10];
    const float* aw   = (const float*)d_in[11];
    const float* ar   = (const float*)d_in[12];
    const float* zci  = (const float*)d_in[13];

    float4* ws4 = (float4*)d_ws;               // needs B*T*16 B = 47.2 MB
    float*  out = (float*)d_out;

    const int tiles  = (B_DIM * T_DIM) / 16;   // 184320, divisible by 8
    prep_kernel<<<tiles / 8, 256, 0, stream>>>(X, W1, b1, W2, b2,
                                               hvac, ig, dg, ws4);
    scan_kernel<<<B_DIM / 128, 128, 0, stream>>>(X, rcR, rcC, winR,
                                                 aw, ar, zci, ws4, out);
}